// CharRNN_30597347017001
// MI455X (gfx1250) — compile-verified
//
#include <hip/hip_runtime.h>
#include <cmath>

typedef __attribute__((ext_vector_type(2))) float v2f;
typedef __attribute__((ext_vector_type(8))) float v8f;

#define B_  64
#define T_  50
#define H_  256
#define V_  50000
#define G_  1024   // 4*H

// ---------------------------------------------------------------------------
// Embedding gather: x_t[(t*B+b)*H + h] = embedding[idx[b*T+t]][h]
// ---------------------------------------------------------------------------
__global__ void embed_gather_kernel(const int* __restrict__ idx,
                                    const float* __restrict__ emb,
                                    float* __restrict__ x_t) {
    int tb = blockIdx.x;            // t*B + b
    int t  = tb / B_;
    int b  = tb - t * B_;
    int v  = idx[b * T_ + t];
    x_t[(size_t)tb * H_ + threadIdx.x] = emb[(size_t)v * H_ + threadIdx.x];
}

// ---------------------------------------------------------------------------
// Generic WMMA f32 GEMM: C[M,N] = A[M,K] * B (+bias)
//   B_TRANS=false: B is [K,N] row-major (row stride ldb)
//   B_TRANS=true : B is [N,K] row-major (row stride ldb)  => C = A * B^T
// One wave computes NT consecutive 16x16 tiles in N. Block = 8 waves.
// ---------------------------------------------------------------------------
template <int NT, bool B_TRANS, bool ADD_BIAS>
__global__ void wmma_gemm_kernel(const float* __restrict__ A, int lda,
                                 const float* __restrict__ B, int ldb,
                                 const float* __restrict__ bias,
                                 float* __restrict__ C, int ldc,
                                 int N, int K) {
    const int wave  = threadIdx.x >> 5;
    const int lane  = threadIdx.x & 31;
    const int m0    = blockIdx.y * 16;
    const int n0    = (blockIdx.x * (blockDim.x >> 5) + wave) * (16 * NT);
    const int mrow  = lane & 15;
    const int khalf = (lane >> 4) << 1;   // 0 for lanes 0-15, 2 for lanes 16-31

    v8f acc[NT];
#pragma unroll
    for (int j = 0; j < NT; ++j) acc[j] = (v8f){0.f,0.f,0.f,0.f,0.f,0.f,0.f,0.f};

    const float* Arow = A + (size_t)(m0 + mrow) * lda + khalf;
    const float* Bp[NT];
#pragma unroll
    for (int j = 0; j < NT; ++j) {
        int n = n0 + j * 16 + (lane & 15);
        if (n > N - 1) n = N - 1;                           // clamp loads, no EXEC change
        Bp[j] = B_TRANS ? (B + (size_t)n * ldb + khalf)
                        : (B + (size_t)khalf * ldb + n);
    }

    for (int kb = 0; kb < K; kb += 4) {
        v2f a;
        a.x = Arow[kb];
        a.y = Arow[kb + 1];
#pragma unroll
        for (int j = 0; j < NT; ++j) {
            v2f b;
            if (B_TRANS) {
                b.x = Bp[j][kb];
                b.y = Bp[j][kb + 1];
            } else {
                b.x = Bp[j][(size_t)kb * ldb];
                b.y = Bp[j][(size_t)(kb + 1) * ldb];
            }
            acc[j] = __builtin_amdgcn_wmma_f32_16x16x4_f32(
                false, a, false, b, (short)0, acc[j], false, false);
        }
    }

    const int rbase = m0 + ((lane >> 4) << 3);   // M = rbase + r
#pragma unroll
    for (int j = 0; j < NT; ++j) {
        int n = n0 + j * 16 + (lane & 15);
        if (n < N) {
            float bb = ADD_BIAS ? bias[n] : 0.0f;
#pragma unroll
            for (int r = 0; r < 8; ++r)
                C[(size_t)(rbase + r) * ldc + n] = acc[j][r] + bb;
        }
    }
}

// ---------------------------------------------------------------------------
// One MI-LSTM timestep for one layer.
// Block = 128 threads = 4 waves (wave w handles gate w: 0=i,1=j,2=f,3=o).
// Each block owns a 16(batch) x 16(hidden) tile:
//   hh = h_prev @ Uh   (WMMA, K=256)
//   gates = alpha*xh*hh + beta1*xh + beta2*hh + bias   (xh precomputed)
//   c = c*sig(f+1) + sig(i)*tanh(j);  h = tanh(c)*sig(o)
// ---------------------------------------------------------------------------
__global__ void mi_lstm_step_kernel(const float* __restrict__ XWt,   // [B,G] for this t
                                    const float* __restrict__ Uh,    // [H,G]
                                    const float* __restrict__ alpha,
                                    const float* __restrict__ beta1,
                                    const float* __restrict__ beta2,
                                    const float* __restrict__ bias,
                                    const float* __restrict__ hprev, int hprev_stride,
                                    float* __restrict__ hout, int hout_stride,
                                    float* __restrict__ c) {
    const int wave  = threadIdx.x >> 5;
    const int lane  = threadIdx.x & 31;
    const int n0    = blockIdx.x * 16;                  // hidden tile
    const int b0    = blockIdx.y * 16;                  // batch tile
    const int gcol  = wave * H_ + n0 + (lane & 15);     // column in G for this gate
    const int khalf = (lane >> 4) << 1;
    const int mrow  = lane & 15;

    v8f acc = (v8f){0.f,0.f,0.f,0.f,0.f,0.f,0.f,0.f};
    const float* Arow = hprev + (size_t)(b0 + mrow) * hprev_stride + khalf;
    const float* Bp   = Uh + (size_t)khalf * G_ + gcol;
    for (int kb = 0; kb < H_; kb += 4) {
        v2f a; a.x = Arow[kb];                 a.y = Arow[kb + 1];
        v2f b; b.x = Bp[(size_t)kb * G_];      b.y = Bp[(size_t)(kb + 1) * G_];
        acc = __builtin_amdgcn_wmma_f32_16x16x4_f32(
            false, a, false, b, (short)0, acc, false, false);
    }

    const float al = alpha[gcol], bt1 = beta1[gcol], bt2 = beta2[gcol], bs = bias[gcol];
    __shared__ float gsm[4][16][16];
    const int rb = (lane >> 4) << 3;
#pragma unroll
    for (int r = 0; r < 8; ++r) {
        int b = b0 + rb + r;
        float xh = XWt[(size_t)b * G_ + gcol];
        float hh = acc[r];
        float g  = al * xh * hh + bt1 * xh + bt2 * hh + bs;
        float v;
        if (wave == 1)      v = tanhf(g);                              // j
        else if (wave == 2) v = 1.0f / (1.0f + __expf(-(g + 1.0f)));   // f + FORGET_BIAS
        else                v = 1.0f / (1.0f + __expf(-g));            // i / o
        gsm[wave][rb + r][lane & 15] = v;
    }
    __syncthreads();

    for (int e = threadIdx.x; e < 256; e += 128) {
        int bb = e >> 4, nn = e & 15;
        int b = b0 + bb, n = n0 + nn;
        float ig = gsm[0][bb][nn], jg = gsm[1][bb][nn];
        float fg = gsm[2][bb][nn], og = gsm[3][bb][nn];
        float cn = c[(size_t)b * H_ + n] * fg + ig * jg;
        c[(size_t)b * H_ + n] = cn;
        hout[(size_t)b * hout_stride + n] = tanhf(cn) * og;
    }
}

// ---------------------------------------------------------------------------
extern "C" void kernel_launch(void* const* d_in, const int* in_sizes, int n_in,
                              void* d_out, int out_size, void* d_ws, size_t ws_size,
                              hipStream_t stream) {
    (void)in_sizes; (void)n_in; (void)out_size; (void)ws_size;

    const int*   idx   = (const int*)d_in[0];
    const float* emb   = (const float*)d_in[1];
    const float* Wx    = (const float*)d_in[2];   // [2,H,G]
    const float* Uh    = (const float*)d_in[3];   // [2,H,G]
    const float* alpha = (const float*)d_in[4];   // [2,G]
    const float* beta1 = (const float*)d_in[5];
    const float* beta2 = (const float*)d_in[6];
    const float* bias  = (const float*)d_in[7];
    const float* sw    = (const float*)d_in[8];   // [V,H]
    const float* sb    = (const float*)d_in[9];   // [V]
    float* out = (float*)d_out;

    const int TB = T_ * B_;                               // 3200
    float* XW     = (float*)d_ws;                         // [TB,G]  x/h projections
    float* bufB   = XW + (size_t)TB * G_;                 // [TB,H]  x_t -> h0_all -> flat
    float* c_buf  = bufB + (size_t)TB * H_;               // [B,H]
    float* h_zero = c_buf + B_ * H_;                      // [B,H]

    hipMemsetAsync(c_buf,  0, B_ * H_ * sizeof(float), stream);
    hipMemsetAsync(h_zero, 0, B_ * H_ * sizeof(float), stream);

    // 1) embedding gather (time-major)
    embed_gather_kernel<<<TB, H_, 0, stream>>>(idx, emb, bufB);

    // 2) XW0 = x_t @ Wx[0]   [3200,256]x[256,1024]
    dim3 gXW(G_ / 512, TB / 16);                          // (2, 200)
    wmma_gemm_kernel<4, false, false><<<gXW, 256, 0, stream>>>(
        bufB, H_, Wx, G_, nullptr, XW, G_, G_, H_);

    // 3) layer-0 recurrence; h0(t) written time-major into bufB
    dim3 gstep(H_ / 16, B_ / 16);                         // (16, 4)
    for (int t = 0; t < T_; ++t) {
        const float* hp = (t == 0) ? h_zero : (bufB + (size_t)(t - 1) * B_ * H_);
        mi_lstm_step_kernel<<<gstep, 128, 0, stream>>>(
            XW + (size_t)t * B_ * G_, Uh, alpha, beta1, beta2, bias,
            hp, H_, bufB + (size_t)t * B_ * H_, H_, c_buf);
    }

    // 4) XW1 = h0_all @ Wx[1]
    wmma_gemm_kernel<4, false, false><<<gXW, 256, 0, stream>>>(
        bufB, H_, Wx + (size_t)H_ * G_, G_, nullptr, XW, G_, G_, H_);

    hipMemsetAsync(c_buf, 0, B_ * H_ * sizeof(float), stream);

    // 5) layer-1 recurrence; h1(t) written batch-major (row b*T+t) into bufB
    const float* Uh1 = Uh + (size_t)H_ * G_;
    for (int t = 0; t < T_; ++t) {
        const float* hp; int hps;
        if (t == 0) { hp = h_zero;                        hps = H_;      }
        else        { hp = bufB + (size_t)(t - 1) * H_;   hps = T_ * H_; }
        mi_lstm_step_kernel<<<gstep, 128, 0, stream>>>(
            XW + (size_t)t * B_ * G_, Uh1,
            alpha + G_, beta1 + G_, beta2 + G_, bias + G_,
            hp, hps, bufB + (size_t)t * H_, T_ * H_, c_buf);
    }

    // 6) logits = flat @ softmax_w^T + softmax_b   [3200,256]x[256,50000]
    dim3 glog((V_ + 511) / 512, TB / 16);                 // (98, 200)
    wmma_gemm_kernel<4, true, true><<<glog, 256, 0, stream>>>(
        bufB, H_, sw, H_, sb, out, V_, V_, H_);
}